// SampleAndGroup_28475633173176
// MI455X (gfx1250) — compile-verified
//
#include <hip/hip_runtime.h>

// Problem constants (match reference setup_inputs)
#define B_  4
#define N_  16384
#define C_  128
#define S_  1024      // SAMPLE_NUM
#define K_  32        // NEIGHBOR_NUM
#define RAD 0.2f

typedef float v2f __attribute__((ext_vector_type(2)));
typedef float v8f __attribute__((ext_vector_type(8)));

// ---------------------------------------------------------------------------
// Kernel 1: Farthest-point sampling. One 1024-thread workgroup per batch.
// Entire batch point cloud (192 KB) staged in CDNA5's 320 KB WGP LDS using
// GLOBAL_LOAD_ASYNC_TO_LDS_B128 (ASYNCcnt-tracked, no VGPR staging); min-dist
// state lives in registers (16 floats/thread). Argmax per step = wave32
// shfl_xor reduce + 32-entry LDS tree.
// ---------------------------------------------------------------------------
__global__ __launch_bounds__(1024) void fps_kernel(const float* __restrict__ xyz,
                                                   float* __restrict__ ws_sxyz,
                                                   float* __restrict__ out_sxyz) {
    __shared__ __align__(16) float lxyz[N_ * 3];   // 192 KB — CDNA5 WGP LDS
    __shared__ float rv[32];
    __shared__ int   ri[32];
    __shared__ int   bcast;

    const int b    = blockIdx.x;
    const int tid  = threadIdx.x;
    const int lane = tid & 31;
    const int wid  = tid >> 5;
    const float* bx = xyz + (size_t)b * N_ * 3;

    // Async copy global -> LDS: 12288 x b128 transfers spread over 1024 lanes.
    // Per-lane LDS offset + per-lane 64-bit global address (GV mode).
    for (int i = tid * 4; i < N_ * 3; i += 1024 * 4) {
        unsigned lds_off = (unsigned)(size_t)(&lxyz[i]);
        unsigned long long ga = (unsigned long long)(const void*)(bx + i);
        asm volatile("global_load_async_to_lds_b128 %0, %1, off"
                     :: "v"(lds_off), "v"(ga) : "memory");
    }
    asm volatile("s_wait_asynccnt 0x0" ::: "memory");  // my wave's copies done
    __syncthreads();                                   // publish to all waves

    const int PPT = N_ / 1024;       // 16 points per thread
    float mind[PPT];
#pragma unroll
    for (int k = 0; k < PPT; k++) mind[k] = 3.4e38f;

    if (tid == 0) {
        unsigned h = 42u * 2654435761u + (unsigned)b * 40503u;  // deterministic start
        bcast = (int)(h % N_);
    }
    __syncthreads();
    int last = bcast;

    for (int s = 0; s < S_; s++) {
        const float lx = lxyz[last * 3 + 0];
        const float ly = lxyz[last * 3 + 1];
        const float lz = lxyz[last * 3 + 2];
        if (tid == 0) {
            size_t o = ((size_t)b * S_ + s) * 3;
            ws_sxyz[o + 0] = lx; ws_sxyz[o + 1] = ly; ws_sxyz[o + 2] = lz;
            out_sxyz[o + 0] = lx; out_sxyz[o + 1] = ly; out_sxyz[o + 2] = lz;
        }
        float bv = -1.0f; int bi = 0;
#pragma unroll
        for (int k = 0; k < PPT; k++) {
            const int n = tid + (k << 10);
            const float dx = lxyz[n * 3 + 0] - lx;
            const float dy = lxyz[n * 3 + 1] - ly;
            const float dz = lxyz[n * 3 + 2] - lz;
            const float d2 = dx * dx + dy * dy + dz * dz;
            mind[k] = fminf(mind[k], d2);
            if (mind[k] > bv) { bv = mind[k]; bi = n; }
        }
        // wave32 (max val, min idx) reduce
#pragma unroll
        for (int off = 16; off >= 1; off >>= 1) {
            const float ov = __shfl_xor(bv, off, 32);
            const int   oi = __shfl_xor(bi, off, 32);
            if (ov > bv || (ov == bv && oi < bi)) { bv = ov; bi = oi; }
        }
        if (lane == 0) { rv[wid] = bv; ri[wid] = bi; }
        __syncthreads();
        if (wid == 0) {
            bv = rv[lane]; bi = ri[lane];
#pragma unroll
            for (int off = 16; off >= 1; off >>= 1) {
                const float ov = __shfl_xor(bv, off, 32);
                const int   oi = __shfl_xor(bi, off, 32);
                if (ov > bv || (ov == bv && oi < bi)) { bv = ov; bi = oi; }
            }
            if (lane == 0) bcast = bi;
        }
        __syncthreads();
        last = bcast;
    }
}

// ---------------------------------------------------------------------------
// Kernel 2: Ball query via V_WMMA_F32_16X16X4_F32.
// A (16x4): row m = (-2qx, -2qy, -2qz, 1); B (4x16): col n = (px, py, pz, |p|^2)
//   => C[m][n] = |p_n|^2 - 2 q_m . p_n, compare against R^2 - |q_m|^2.
// One wave per 16-query tile; up to 1024 WMMAs sweep all N points with the
// next tile's point coordinates software-pipelined past the current WMMA.
// Ballot per C register yields 16-bit pass masks for rows r (low) / r+8
// (high); lanes 0..15 own one query row and append first-K indices in index
// order. Padding uses sentinel index S_ (ref: idx==-1 -> gather feat[:,1024,:]).
// ---------------------------------------------------------------------------
__global__ __launch_bounds__(32) void ballq_kernel(const float* __restrict__ xyz,
                                                   const float* __restrict__ sxyz,
                                                   int* __restrict__ nidx) {
    const int b    = blockIdx.x >> 6;      // 64 query-tiles per batch (1024/16)
    const int tq   = blockIdx.x & 63;
    const int s0   = tq << 4;
    const int lane = threadIdx.x;
    const int hi   = lane >> 4;
    const int m    = lane & 15;
    const float R2 = RAD * RAD;
    const int NT   = N_ / 16;

    const float* bx = xyz  + (size_t)b * N_ * 3;
    const float* qs = sxyz + ((size_t)b * S_ + s0) * 3;

    // A-matrix operand: VGPR0 = K0/K2, VGPR1 = K1/K3 (lanes 0-15 / 16-31)
    const float qx = qs[m * 3 + 0], qy = qs[m * 3 + 1], qz = qs[m * 3 + 2];
    v2f a;
    a.x = hi ? (-2.0f * qz) : (-2.0f * qx);
    a.y = hi ? 1.0f         : (-2.0f * qy);

    // Per-lane thresholds for the C rows this lane group holds (r + 8*hi)
    float thr[8];
#pragma unroll
    for (int r = 0; r < 8; r++) {
        const int row = r + (hi << 3);
        const float x = qs[row * 3 + 0], y = qs[row * 3 + 1], z = qs[row * 3 + 2];
        thr[r] = R2 - (x * x + y * y + z * z);
    }

    int cnt = (lane < 16) ? 0 : K_;        // hi lanes own no row -> "done"
    int* outrow = nidx + ((size_t)b * S_ + (s0 + m)) * K_;

    // Pipeline prologue: load tile 0's point for this lane.
    float px = bx[m * 3 + 0], py = bx[m * 3 + 1], pz = bx[m * 3 + 2];

    for (int tile = 0; tile < NT; tile++) {
        // Issue next tile's load early (clamped on last tile); overlaps WMMA.
        const int nn = ((tile + 1 < NT) ? (tile + 1) : tile) * 16 + m;
        const float fx = bx[nn * 3 + 0];
        const float fy = bx[nn * 3 + 1];
        const float fz = bx[nn * 3 + 2];

        const int nb = tile << 4;
        v2f bmat;
        bmat.x = hi ? pz : px;
        bmat.y = hi ? (px * px + py * py + pz * pz) : py;

        v8f c = {0.f, 0.f, 0.f, 0.f, 0.f, 0.f, 0.f, 0.f};
        c = __builtin_amdgcn_wmma_f32_16x16x4_f32(false, a, false, bmat,
                                                  (short)0, c, false, false);
        unsigned msk[8];
#pragma unroll
        for (int r = 0; r < 8; r++)
            msk[r] = __builtin_amdgcn_ballot_w32(c[r] < thr[r]);

        if (lane < 16) {
            unsigned mm = (m < 8) ? (msk[m] & 0xFFFFu) : (msk[m - 8] >> 16);
            while (mm && cnt < K_) {
                const int bit = __builtin_ctz(mm);
                outrow[cnt++] = nb + bit;
                mm &= mm - 1;
            }
        }
        if (__builtin_amdgcn_ballot_w32(cnt < K_) == 0u) break;   // all rows full
        px = fx; py = fy; pz = fz;
    }
    if (lane < 16)
        for (; cnt < K_; cnt++) outrow[cnt] = S_;  // ref pad semantics
}

// ---------------------------------------------------------------------------
// Kernel 3: Feature gather. One wave per output row (b,s,k): 32 lanes x
// float4 = 128 floats = one b128 load + one b128 store per lane. This is the
// only bandwidth-significant stage (~64 MB of stores at 23.3 TB/s).
// ---------------------------------------------------------------------------
__global__ __launch_bounds__(256) void gather_kernel(const float* __restrict__ feat,
                                                     const int* __restrict__ nidx,
                                                     float* __restrict__ outf) {
    const int row  = (blockIdx.x << 3) + (int)(threadIdx.x >> 5);
    const int lane = threadIdx.x & 31;
    const int b    = row >> 15;            // S_*K_ = 32768 rows per batch
    const int n    = nidx[row];
    const float4* src = (const float4*)(feat + (((size_t)b * N_ + n) * C_)) + lane;
    float4*       dst = (float4*)(outf + ((size_t)row * C_)) + lane;
    *dst = *src;
}

extern "C" void kernel_launch(void* const* d_in, const int* in_sizes, int n_in,
                              void* d_out, int out_size, void* d_ws, size_t ws_size,
                              hipStream_t stream) {
    (void)in_sizes; (void)n_in; (void)out_size; (void)ws_size;
    const float* xyz  = (const float*)d_in[0];
    const float* feat = (const float*)d_in[1];

    float* out       = (float*)d_out;
    float* out_sxyz  = out;                              // (B,S,3)
    float* out_feat  = out + (size_t)B_ * S_ * 3;        // (B,S,K,C)

    float* ws_sxyz = (float*)d_ws;
    size_t off = (((size_t)B_ * S_ * 3 * sizeof(float)) + 255) & ~(size_t)255;
    int*   ws_idx  = (int*)((char*)d_ws + off);

    fps_kernel  <<<B_,              1024, 0, stream>>>(xyz, ws_sxyz, out_sxyz);
    ballq_kernel<<<B_ * (S_ / 16),    32, 0, stream>>>(xyz, ws_sxyz, ws_idx);
    gather_kernel<<<(B_ * S_ * K_) / 8, 256, 0, stream>>>(feat, ws_idx, out_feat);
}